// SimplePendulumSolver_13546326852239
// MI455X (gfx1250) — compile-verified
//
#include <hip/hip_runtime.h>
#include <hip/hip_bf16.h>
#include <stdint.h>

// ---------------------------------------------------------------------------
// Serial RK4 pendulum integration. Latency-bound: one wave, one logical lane.
// CDNA5 paths used: Tensor Data Mover (tensor_load_to_lds) double-buffered
// staging of the time grid into LDS, S_WAIT_TENSORCNT for completion.
// WMMA is inapplicable (nonlinear scalar recurrence, no matmul structure).
// ---------------------------------------------------------------------------

typedef unsigned int u32x4 __attribute__((ext_vector_type(4)));
typedef int          i32x4 __attribute__((ext_vector_type(4)));
typedef int          i32x8 __attribute__((ext_vector_type(8)));

#define CHUNK 4096   // floats per TDM chunk (16 KB); 2 buffers = 32 KB LDS

// Issue one TDM load: `tile` contiguous f32 elements from gptr -> LDS offset.
// `avail` = valid elements remaining from gptr (OOB reads beyond it return 0).
__device__ __forceinline__ void tdm_load_chunk(unsigned lds_off,
                                               const float* gptr,
                                               unsigned avail,
                                               unsigned tile) {
  unsigned long long ga = (unsigned long long)(uintptr_t)gptr;

  // D# group 0 (128b): count=1 | lds_addr | global_addr[56:0] | type=2
  u32x4 g0;
  g0.x = 1u;                                              // count=1 (valid)
  g0.y = lds_off;                                         // lds_addr (bytes)
  g0.z = (unsigned)(ga & 0xFFFFFFFFull);                  // global_addr[31:0]
  g0.w = (unsigned)((ga >> 32) & 0x01FFFFFFull)           // global_addr[56:32]
         | 0x80000000u;                                   // type=2 ("image")

  // D# group 1 (256b): data_size=4B, tensor_dim0=avail, tensor_dim1=1,
  // tile_dim0=tile (1D tile), dim0_stride=avail, everything else 0.
  i32x8 g1;
  g1[0] = (int)(2u << 16);                                // data_size: 4 bytes
  g1[1] = (int)((avail & 0xFFFFu) << 16);                 // tensor_dim0[15:0]
  g1[2] = (int)((avail >> 16) | (1u << 16));              // dim0[31:16] | dim1=1
  g1[3] = (int)(tile << 16);                              // tile_dim0
  g1[4] = 0;                                              // tile_dim1/2 unused
  g1[5] = (int)avail;                                     // dim0_stride[31:0]
  g1[6] = 0;
  g1[7] = 0;

  i32x4 z4 = {0, 0, 0, 0};
#if __clang_major__ >= 23
  i32x8 z8 = {0, 0, 0, 0, 0, 0, 0, 0};
  __builtin_amdgcn_tensor_load_to_lds(g0, g1, z4, z4, z8, 0);
#else
  __builtin_amdgcn_tensor_load_to_lds(g0, g1, z4, z4, 0);
#endif
}

__device__ __forceinline__ void rk4_step(float& a, float& b, float dt, float om) {
  // y = [theta, theta_dot]; dy = [theta_dot, -om*sin(theta)]
  float k1a = b;
  float k1b = -om * __sinf(a);
  float a2 = a + 0.5f * dt * k1a, b2 = b + 0.5f * dt * k1b;
  float k2a = b2;
  float k2b = -om * __sinf(a2);
  float a3 = a + 0.5f * dt * k2a, b3 = b + 0.5f * dt * k2b;
  float k3a = b3;
  float k3b = -om * __sinf(a3);
  float a4 = a + dt * k3a, b4 = b + dt * k3b;
  float k4a = b4;
  float k4b = -om * __sinf(a4);
  float s = dt * (1.0f / 6.0f);
  a += s * (k1a + 2.0f * k2a + 2.0f * k3a + k4a);
  b += s * (k1b + 2.0f * k2b + 2.0f * k3b + k4b);
}

__global__ __launch_bounds__(32)
void pendulum_rk4_kernel(const float* __restrict__ y0,
                         const float* __restrict__ t_span,
                         const float* __restrict__ omega,
                         float* __restrict__ out,
                         int n) {
  __shared__ float sdt[2][CHUNK];

  const int total = n - 1;                       // number of RK4 steps
  const int nchunks = (total + CHUNK - 1) / CHUNK;
  const bool lane0 = (threadIdx.x == 0);

  // Uniform scalar state (all lanes redundantly compute; SIMD makes it free).
  float a  = y0[0];
  float b  = y0[1];
  float om = omega[0];
  float tprev = t_span[0];

  if (lane0) out[0] = a;

  // LDS byte offsets of the two staging buffers.
  unsigned lds0 = (unsigned)(uintptr_t)&sdt[0][0];
  unsigned lds1 = (unsigned)(uintptr_t)&sdt[1][0];

  // Prime the pipeline: chunk 0 holds t_span[1 .. CHUNK].
  tdm_load_chunk(lds0, t_span + 1, (unsigned)(n - 1), CHUNK);

  for (int c = 0; c < nchunks; ++c) {
    // Kick off the next chunk's DMA, then wait until the current one landed.
    if (c + 1 < nchunks) {
      int nbase = (c + 1) * CHUNK + 1;
      tdm_load_chunk(((c + 1) & 1) ? lds1 : lds0,
                     t_span + nbase, (unsigned)(n - nbase), CHUNK);
      __builtin_amdgcn_s_wait_tensorcnt(1);   // in-order: oldest (chunk c) done
    } else {
      __builtin_amdgcn_s_wait_tensorcnt(0);
    }

    const float* ts = sdt[c & 1];
    const int base = c * CHUNK;
    const int len  = (total - base < CHUNK) ? (total - base) : CHUNK;

    for (int j = 0; j < len; ++j) {
      float tcur = ts[j];                 // LDS load, off the critical chain
      float dt = tcur - tprev;            // matches jnp.diff(t_span) exactly
      tprev = tcur;
      rk4_step(a, b, dt, om);
      if (lane0) out[base + j + 1] = a;   // fire-and-forget store
    }
  }
}

extern "C" void kernel_launch(void* const* d_in, const int* in_sizes, int n_in,
                              void* d_out, int out_size, void* d_ws, size_t ws_size,
                              hipStream_t stream) {
  (void)n_in; (void)out_size; (void)d_ws; (void)ws_size;
  const float* y0    = (const float*)d_in[0];   // (2,)
  const float* tspan = (const float*)d_in[1];   // (65536,)
  const float* omega = (const float*)d_in[2];   // (1,)
  float* out = (float*)d_out;                   // (65536,)
  int n = in_sizes[1];

  // Inherently serial scan: one block, one wave.
  pendulum_rk4_kernel<<<dim3(1), dim3(32), 0, stream>>>(y0, tspan, omega, out, n);
}